// TumorGCNClassifier_22230750724496
// MI455X (gfx1250) — compile-verified
//
#include <hip/hip_runtime.h>

// Problem constants (from reference)
#define NN   100000
#define EE   800000
#define GG   64
#define FIN  64
#define HH   128
#define CC   4

typedef __attribute__((ext_vector_type(2))) float v2f;
typedef __attribute__((ext_vector_type(8))) float v8f;

// ---------------------------------------------------------------------------
// Degree / normalization
// ---------------------------------------------------------------------------
__global__ void k_deg_init(float* __restrict__ deg) {
    int i = blockIdx.x * blockDim.x + threadIdx.x;
    if (i < NN) deg[i] = 1.0f;                       // self-loop
}

__global__ void k_deg_count(const int* __restrict__ dst, float* __restrict__ deg) {
    int e = blockIdx.x * blockDim.x + threadIdx.x;
    if (e < EE) atomicAdd(&deg[dst[e]], 1.0f);
}

__global__ void k_dinv(float* __restrict__ deg) {
    int i = blockIdx.x * blockDim.x + threadIdx.x;
    if (i < NN) deg[i] = rsqrtf(deg[i]);             // in-place: deg -> dinv
}

// ---------------------------------------------------------------------------
// Weight repack: B[K,Ncol] row-major -> WMMA B-fragment order, so the GEMM
// inner loop loads one coalesced b64 per fragment instead of 2 strided b32.
// Bp[((kb * (Ncol/16) + ct) * 32 + lane)] = { B[kb*4+kg][ct*16+m],
//                                             B[kb*4+kg+1][ct*16+m] }
//   with m = lane&15, kg = 2*(lane>>4)
// ---------------------------------------------------------------------------
__global__ void k_pack_B(const float* __restrict__ B, v2f* __restrict__ Bp,
                         int K, int Ncol) {
    int idx = blockIdx.x * blockDim.x + threadIdx.x;
    int total = (K >> 2) * (Ncol >> 4) * 32;
    if (idx < total) {
        int lane = idx & 31;
        int t  = idx >> 5;
        int ct = t % (Ncol >> 4);
        int kb = t / (Ncol >> 4);
        int m  = lane & 15;
        int k  = kb * 4 + ((lane >> 4) << 1);
        v2f v;
        v[0] = B[(size_t)k       * Ncol + ct * 16 + m];
        v[1] = B[(size_t)(k + 1) * Ncol + ct * 16 + m];
        Bp[idx] = v;
    }
}

// ---------------------------------------------------------------------------
// fp32 WMMA GEMM: D[M,Ncol] = A[M,K] @ B[K,Ncol], B pre-packed.
// One wave computes a 32x32 output block as a 2x2 grid of 16x16 WMMA tiles:
// 4 v_wmma per K-step against 2 A-fragment b64 loads + 2 packed B b64 loads.
// Grids are exact (M,Ncol multiples of 32) -> EXEC all-ones at every WMMA.
// A frag (16x4): lane m = lane&15, K pair = 2*(lane>>4) + {0,1}
// D frag (16x16): row = 8*(lane>>4) + v, col = lane&15
// ---------------------------------------------------------------------------
__global__ void k_gemm_wmma(const float* __restrict__ A, const v2f* __restrict__ Bp,
                            float* __restrict__ D, int M, int K, int Ncol) {
    const int lane = threadIdx.x & 31;
    const int wave = threadIdx.x >> 5;
    const int colGroups = Ncol >> 5;                 // 32-wide column groups
    const int nBlocks32 = (M >> 5) * colGroups;
    const int tile = blockIdx.x * (blockDim.x >> 5) + wave;
    if (tile >= nBlocks32) return;                   // wave-uniform branch

    const int row0 = (tile / colGroups) << 5;
    const int colG = tile % colGroups;
    const int ct0  = colG << 1;                      // first 16-col tile index
    const int tpc  = Ncol >> 4;                      // 16-col tiles per row
    const int m    = lane & 15;
    const int kg   = (lane >> 4) << 1;

    const float* a0p = A + (size_t)(row0 + m) * K + kg;
    const float* a1p = a0p + (size_t)16 * K;

    v8f acc00 = {}, acc01 = {}, acc10 = {}, acc11 = {};
    for (int k = 0; k < K; k += 4) {
        v2f a0 = *(const v2f*)(a0p + k);
        v2f a1 = *(const v2f*)(a1p + k);
        const v2f* bp = Bp + ((size_t)(k >> 2) * tpc + ct0) * 32 + lane;
        v2f b0 = bp[0];
        v2f b1 = bp[32];
        acc00 = __builtin_amdgcn_wmma_f32_16x16x4_f32(false, a0, false, b0,
                                                      (short)0, acc00, false, false);
        acc01 = __builtin_amdgcn_wmma_f32_16x16x4_f32(false, a0, false, b1,
                                                      (short)0, acc01, false, false);
        acc10 = __builtin_amdgcn_wmma_f32_16x16x4_f32(false, a1, false, b0,
                                                      (short)0, acc10, false, false);
        acc11 = __builtin_amdgcn_wmma_f32_16x16x4_f32(false, a1, false, b1,
                                                      (short)0, acc11, false, false);
    }

    const int rb   = (lane >> 4) << 3;               // D-frag row base within tile
    const int col0 = colG << 5;
    float* d0 = D + (size_t)(row0 + rb) * Ncol + col0 + m;
    float* d1 = d0 + (size_t)16 * Ncol;
#pragma unroll
    for (int v = 0; v < 8; ++v) {
        d0[(size_t)v * Ncol]      = acc00[v];
        d0[(size_t)v * Ncol + 16] = acc01[v];
        d1[(size_t)v * Ncol]      = acc10[v];
        d1[(size_t)v * Ncol + 16] = acc11[v];
    }
}

// ---------------------------------------------------------------------------
// GCN aggregation pieces
// ---------------------------------------------------------------------------
// agg = h * dinv^2 + bias   (self-loop term, bias folded in before scatter)
__global__ void k_self_init(const float* __restrict__ h, const float* __restrict__ dinv,
                            const float* __restrict__ bias, float* __restrict__ agg) {
    int i = blockIdx.x * blockDim.x + threadIdx.x;   // over N*H
    if (i < NN * HH) {
        int node = i >> 7;                           // /HH
        int f    = i & (HH - 1);
        float di = dinv[node];
        agg[i] = h[i] * di * di + bias[f];
    }
}

// One wave per edge; each lane moves 4 floats (float4 gather + 4 f32 atomics)
__global__ void k_edge_scatter(const float* __restrict__ h,
                               const int* __restrict__ src, const int* __restrict__ dst,
                               const float* __restrict__ dinv, float* __restrict__ agg) {
    int idx = blockIdx.x * blockDim.x + threadIdx.x; // over E * (H/4)
    int e  = idx >> 5;                               // H/4 == 32 lanes per edge
    int f4 = (idx & 31) << 2;
    if (e < EE) {
        int s = src[e], d = dst[e];
        float nrm = dinv[s] * dinv[d];
        float4 hv = *(const float4*)(h + (size_t)s * HH + f4);
        float* ap = agg + (size_t)d * HH + f4;
        atomicAdd(ap + 0, hv.x * nrm);
        atomicAdd(ap + 1, hv.y * nrm);
        atomicAdd(ap + 2, hv.z * nrm);
        atomicAdd(ap + 3, hv.w * nrm);
    }
}

__global__ void k_relu(float* __restrict__ a, int n) {
    int i = blockIdx.x * blockDim.x + threadIdx.x;
    if (i < n) a[i] = fmaxf(a[i], 0.0f);
}

// ---------------------------------------------------------------------------
// Global mean pool + MLP head
// ---------------------------------------------------------------------------
__global__ void k_zero(float* __restrict__ a, int n) {
    int i = blockIdx.x * blockDim.x + threadIdx.x;
    if (i < n) a[i] = 0.0f;
}

__global__ void k_pool_scatter(const float* __restrict__ h, const int* __restrict__ batch,
                               float* __restrict__ s) {
    int idx = blockIdx.x * blockDim.x + threadIdx.x; // over N * (H/4)
    int node = idx >> 5;
    int f4   = (idx & 31) << 2;
    if (node < NN) {
        int g = batch[node];
        float4 hv = *(const float4*)(h + (size_t)node * HH + f4);
        float* sp = s + (size_t)g * HH + f4;
        atomicAdd(sp + 0, hv.x);
        atomicAdd(sp + 1, hv.y);
        atomicAdd(sp + 2, hv.z);
        atomicAdd(sp + 3, hv.w);
    }
}

__global__ void k_pool_cnt(const int* __restrict__ batch, float* __restrict__ cnt) {
    int i = blockIdx.x * blockDim.x + threadIdx.x;
    if (i < NN) atomicAdd(&cnt[batch[i]], 1.0f);
}

__global__ void k_pool_div(const float* __restrict__ s, const float* __restrict__ cnt,
                           float* __restrict__ pooled) {
    int i = blockIdx.x * blockDim.x + threadIdx.x;   // over G*H
    if (i < GG * HH) pooled[i] = s[i] / fmaxf(cnt[i >> 7], 1.0f);
}

__global__ void k_bias_relu(float* __restrict__ a, const float* __restrict__ b, int n) {
    int i = blockIdx.x * blockDim.x + threadIdx.x;
    if (i < n) a[i] = fmaxf(a[i] + b[i & (HH - 1)], 0.0f);
}

// out[G,C] = z[G,H] @ fw2[H,C] + fb2  (tiny: one block, 256 threads = 64*4)
__global__ void k_mlp2(const float* __restrict__ z, const float* __restrict__ fw2,
                       const float* __restrict__ fb2, float* __restrict__ out) {
    int idx = threadIdx.x;
    int g = idx >> 2, c = idx & 3;
    float acc = fb2[c];
#pragma unroll 8
    for (int k = 0; k < HH; ++k)
        acc += z[(size_t)g * HH + k] * fw2[(size_t)k * CC + c];
    out[g * CC + c] = acc;
}

// ---------------------------------------------------------------------------
// Host launcher
// ---------------------------------------------------------------------------
extern "C" void kernel_launch(void* const* d_in, const int* in_sizes, int n_in,
                              void* d_out, int out_size, void* d_ws, size_t ws_size,
                              hipStream_t stream) {
    const float* x   = (const float*)d_in[0];
    const float* W1  = (const float*)d_in[1];
    const float* b1  = (const float*)d_in[2];
    const float* W2  = (const float*)d_in[3];
    const float* b2  = (const float*)d_in[4];
    const float* W3  = (const float*)d_in[5];
    const float* b3  = (const float*)d_in[6];
    const float* fw1 = (const float*)d_in[7];
    const float* fb1 = (const float*)d_in[8];
    const float* fw2 = (const float*)d_in[9];
    const float* fb2 = (const float*)d_in[10];
    const int* eidx  = (const int*)d_in[11];
    const int* batch = (const int*)d_in[12];
    const int* src = eidx;          // edge_index[0]
    const int* dst = eidx + EE;     // edge_index[1]
    float* out = (float*)d_out;

    // workspace layout (floats; all region sizes even -> v2f stays 8B aligned)
    float* ws    = (float*)d_ws;
    float* dinv  = ws;                         // N        (deg -> dinv in place)
    float* hbuf  = dinv + NN;                  // N*H
    float* agg   = hbuf + (size_t)NN * HH;     // N*H
    float* psum  = agg + (size_t)NN * HH;      // G*H
    float* pcnt  = psum + GG * HH;             // G (64, keeps alignment even)
    float* pooled= pcnt + GG;                  // G*H
    float* zbuf  = pooled + GG * HH;           // G*H
    v2f*   bpack = (v2f*)(zbuf + GG * HH);     // max (128/4)*(128/16)*32 v2f = 64KB

    const int T = 256;
    const int nhBlocks   = (NN * HH + T - 1) / T;            // 50000
    const int gemmBlocks = ((NN >> 5) * (HH >> 5)) / 4;      // 3125 (4 waves/blk)
    const int edgeBlocks = (EE * (HH / 4) + T - 1) / T;      // 100000
    const int poolBlocks = (NN * (HH / 4) + T - 1) / T;      // 12500
    const int pack64Blk  = ((FIN >> 2) * (HH >> 4) * 32 + T - 1) / T;  // 16
    const int pack128Blk = ((HH >> 2)  * (HH >> 4) * 32 + T - 1) / T;  // 32

    // degrees -> dinv
    k_deg_init <<<(NN + T - 1) / T, T, 0, stream>>>(dinv);
    k_deg_count<<<(EE + T - 1) / T, T, 0, stream>>>(dst, dinv);
    k_dinv     <<<(NN + T - 1) / T, T, 0, stream>>>(dinv);

    // ---- layer 1: x[N,64] -> agg[N,128], ReLU
    k_pack_B      <<<pack64Blk, T, 0, stream>>>(W1, bpack, FIN, HH);
    k_gemm_wmma   <<<gemmBlocks, 128, 0, stream>>>(x, bpack, hbuf, NN, FIN, HH);
    k_self_init   <<<nhBlocks, T, 0, stream>>>(hbuf, dinv, b1, agg);
    k_edge_scatter<<<edgeBlocks, T, 0, stream>>>(hbuf, src, dst, dinv, agg);
    k_relu        <<<nhBlocks, T, 0, stream>>>(agg, NN * HH);

    // ---- layer 2: agg -> agg, ReLU
    k_pack_B      <<<pack128Blk, T, 0, stream>>>(W2, bpack, HH, HH);
    k_gemm_wmma   <<<gemmBlocks, 128, 0, stream>>>(agg, bpack, hbuf, NN, HH, HH);
    k_self_init   <<<nhBlocks, T, 0, stream>>>(hbuf, dinv, b2, agg);
    k_edge_scatter<<<edgeBlocks, T, 0, stream>>>(hbuf, src, dst, dinv, agg);
    k_relu        <<<nhBlocks, T, 0, stream>>>(agg, NN * HH);

    // ---- layer 3: agg -> agg (no ReLU)
    k_pack_B      <<<pack128Blk, T, 0, stream>>>(W3, bpack, HH, HH);
    k_gemm_wmma   <<<gemmBlocks, 128, 0, stream>>>(agg, bpack, hbuf, NN, HH, HH);
    k_self_init   <<<nhBlocks, T, 0, stream>>>(hbuf, dinv, b3, agg);
    k_edge_scatter<<<edgeBlocks, T, 0, stream>>>(hbuf, src, dst, dinv, agg);

    // ---- global mean pool
    k_zero        <<<(GG * HH + GG + T - 1) / T, T, 0, stream>>>(psum, GG * HH + GG);
    k_pool_scatter<<<poolBlocks, T, 0, stream>>>(agg, batch, psum);
    k_pool_cnt    <<<(NN + T - 1) / T, T, 0, stream>>>(batch, pcnt);
    k_pool_div    <<<(GG * HH + T - 1) / T, T, 0, stream>>>(psum, pcnt, pooled);

    // ---- MLP head: pooled[64,128] @ fw1 -> relu -> @ fw2 -> out[64,4]
    k_pack_B    <<<pack128Blk, T, 0, stream>>>(fw1, bpack, HH, HH);
    k_gemm_wmma <<<((GG >> 5) * (HH >> 5) + 3) / 4, 128, 0, stream>>>(pooled, bpack, zbuf, GG, HH, HH);
    k_bias_relu <<<(GG * HH + T - 1) / T, T, 0, stream>>>(zbuf, fb1, GG * HH);
    k_mlp2      <<<1, GG * CC, 0, stream>>>(zbuf, fw2, fb2, out);
}